// DirGCNConv_62723702391593
// MI455X (gfx1250) — compile-verified
//
#include <hip/hip_runtime.h>

#define N_NODES 100000
#define N_EDGES 600000
#define DIM     128
#define ALPHA   0.5f

typedef __attribute__((ext_vector_type(2))) float v2f;
typedef __attribute__((ext_vector_type(8))) float v8f;

// ---------------- kernel 1: zero scratch (deg_out, deg_in, agg_src, agg_dst) ----
__global__ void dirgcn_zero_kernel(float* __restrict__ p, long n) {
    long i = (long)blockIdx.x * blockDim.x + threadIdx.x;
    long stride = (long)gridDim.x * blockDim.x;
    for (; i < n; i += stride) p[i] = 0.0f;
}

// ---------------- kernel 2: degree counts -------------------------------------
__global__ void dirgcn_degree_kernel(const int* __restrict__ ei,
                                     float* __restrict__ deg_out,
                                     float* __restrict__ deg_in) {
    int e = blockIdx.x * blockDim.x + threadIdx.x;
    if (e < N_EDGES) {
        atomicAdd(&deg_out[ei[e]], 1.0f);            // row = edge_index[0][e]
        atomicAdd(&deg_in[ei[N_EDGES + e]], 1.0f);   // col = edge_index[1][e]
    }
}

// ---------------- kernel 3: one wave per edge, weighted scatter-add ------------
__global__ void dirgcn_scatter_kernel(const float* __restrict__ x,
                                      const int* __restrict__ ei,
                                      const float* __restrict__ deg_out,
                                      const float* __restrict__ deg_in,
                                      float* __restrict__ agg_src,
                                      float* __restrict__ agg_dst) {
    int wave = blockIdx.x * (blockDim.x >> 5) + (threadIdx.x >> 5);
    int lane = threadIdx.x & 31;
    int row = ei[wave];
    int col = ei[N_EDGES + wave];
    float dro = deg_out[row];
    float dci = deg_in[col];
    float ws = (dro > 0.0f) ? rsqrtf(fmaxf(dro, 1.0f)) : 0.0f;
    float wd = (dci > 0.0f) ? rsqrtf(fmaxf(dci, 1.0f)) : 0.0f;
    float w = ws * wd;

    const float4* xc = (const float4*)(x + (size_t)col * DIM);
    const float4* xr = (const float4*)(x + (size_t)row * DIM);
    float4 vc = xc[lane];   // 4 floats/lane * 32 lanes = 128
    float4 vr = xr[lane];

    float* as = agg_src + (size_t)row * DIM + lane * 4;
    float* ad = agg_dst + (size_t)col * DIM + lane * 4;
    atomicAdd(as + 0, w * vc.x);
    atomicAdd(as + 1, w * vc.y);
    atomicAdd(as + 2, w * vc.z);
    atomicAdd(as + 3, w * vc.w);
    atomicAdd(ad + 0, w * vr.x);
    atomicAdd(ad + 1, w * vr.y);
    atomicAdd(ad + 2, w * vr.z);
    atomicAdd(ad + 3, w * vr.w);
}

// ---------------- kernel 4: pre-scale weights / fuse bias ----------------------
__global__ void dirgcn_wprep_kernel(const float* __restrict__ Ws,
                                    const float* __restrict__ bs,
                                    const float* __restrict__ Wd,
                                    const float* __restrict__ bd,
                                    float* __restrict__ Ws_s,
                                    float* __restrict__ Wd_s,
                                    float* __restrict__ bias) {
    int i = blockIdx.x * blockDim.x + threadIdx.x;
    if (i < DIM * DIM) {
        Ws_s[i] = (1.0f - ALPHA) * Ws[i];
        Wd_s[i] = ALPHA * Wd[i];
    }
    if (i < DIM) bias[i] = (1.0f - ALPHA) * bs[i] + ALPHA * bd[i];
}

// ---------------- kernel 5: fused dual GEMM via fp32 WMMA ----------------------
// out = agg_src @ Ws_s^T + agg_dst @ Wd_s^T + bias
// 1 block = one 16-node tile; 8 waves = 8 feature tiles (j0 = wave*16), K=128.
__global__ void __launch_bounds__(256)
dirgcn_gemm_kernel(const float* __restrict__ agg_src,
                   const float* __restrict__ agg_dst,
                   const float* __restrict__ Ws,
                   const float* __restrict__ Wd,
                   const float* __restrict__ bias,
                   float* __restrict__ out) {
    const int lane = threadIdx.x & 31;
    const int wave = threadIdx.x >> 5;
    const int row0 = blockIdx.x * 16;   // node tile
    const int j0   = wave * 16;         // output-feature tile
    const int m    = lane & 15;         // M (A) or N (B) index
    const int koff = (lane >> 4) * 2;   // lanes 0-15: K={k,k+1}; lanes 16-31: K={k+2,k+3}

    const float* arow_s = agg_src + (size_t)(row0 + m) * DIM;
    const float* arow_d = agg_dst + (size_t)(row0 + m) * DIM;
    const float* wrow_s = Ws + (size_t)(j0 + m) * DIM;   // B[k][n] = W[n][k]
    const float* wrow_d = Wd + (size_t)(j0 + m) * DIM;

    v8f acc = {};
    #pragma unroll
    for (int k = 0; k < DIM; k += 4) {
        v2f a = *(const v2f*)(arow_s + k + koff);
        v2f b = *(const v2f*)(wrow_s + k + koff);
        acc = __builtin_amdgcn_wmma_f32_16x16x4_f32(false, a, false, b,
                                                    (short)0, acc, false, false);
    }
    #pragma unroll
    for (int k = 0; k < DIM; k += 4) {
        v2f a = *(const v2f*)(arow_d + k + koff);
        v2f b = *(const v2f*)(wrow_d + k + koff);
        acc = __builtin_amdgcn_wmma_f32_16x16x4_f32(false, a, false, b,
                                                    (short)0, acc, false, false);
    }

    float bv = bias[j0 + m];            // n = lane & 15
    const int n = lane & 15;
    const int mbase = (lane >> 4) * 8;  // lanes 0-15: rows 0..7; lanes 16-31: rows 8..15
    #pragma unroll
    for (int r = 0; r < 8; ++r) {
        out[(size_t)(row0 + mbase + r) * DIM + j0 + n] = acc[r] + bv;
    }
}

// ---------------- launcher -----------------------------------------------------
extern "C" void kernel_launch(void* const* d_in, const int* in_sizes, int n_in,
                              void* d_out, int out_size, void* d_ws, size_t ws_size,
                              hipStream_t stream) {
    const float* x    = (const float*)d_in[0];
    const int*   ei   = (const int*)d_in[1];   // [2, E] row-major
    const float* W_src = (const float*)d_in[2];
    const float* b_src = (const float*)d_in[3];
    const float* W_dst = (const float*)d_in[4];
    const float* b_dst = (const float*)d_in[5];
    float* out = (float*)d_out;

    float* ws = (float*)d_ws;
    float* deg_out = ws;                                    // N
    float* deg_in  = deg_out + N_NODES;                     // N
    float* agg_src = deg_in + N_NODES;                      // N*128
    float* agg_dst = agg_src + (size_t)N_NODES * DIM;       // N*128
    float* Ws_s    = agg_dst + (size_t)N_NODES * DIM;       // 128*128
    float* Wd_s    = Ws_s + DIM * DIM;                      // 128*128
    float* bias    = Wd_s + DIM * DIM;                      // 128

    const long n_zero = 2L * N_NODES + 2L * N_NODES * DIM;  // deg + agg, contiguous
    dirgcn_zero_kernel<<<2048, 256, 0, stream>>>(ws, n_zero);

    dirgcn_degree_kernel<<<(N_EDGES + 255) / 256, 256, 0, stream>>>(ei, deg_out, deg_in);

    // 8 waves per 256-thread block -> E/8 = 75000 blocks (exact)
    dirgcn_scatter_kernel<<<N_EDGES / 8, 256, 0, stream>>>(x, ei, deg_out, deg_in,
                                                           agg_src, agg_dst);

    dirgcn_wprep_kernel<<<(DIM * DIM + 255) / 256, 256, 0, stream>>>(
        W_src, b_src, W_dst, b_dst, Ws_s, Wd_s, bias);

    // 100000/16 = 6250 node tiles (exact)
    dirgcn_gemm_kernel<<<N_NODES / 16, 256, 0, stream>>>(agg_src, agg_dst,
                                                         Ws_s, Wd_s, bias, out);
}